// Attention_43447889167091
// MI455X (gfx1250) — compile-verified
//
#include <hip/hip_runtime.h>
#include <hip/hip_bf16.h>

typedef __attribute__((ext_vector_type(2))) float v2f;
typedef __attribute__((ext_vector_type(8))) float v8f;

// ---------------------------------------------------------------------------
// Kernel 1: energy[b,s] = sum_d key[s,b,d] * query[b,d]  via V_WMMA_F32_16X16X4_F32
// Each wave computes a 16-row s-tile for one batch b. A = key tile (16x4),
// B = query[k0..k0+3] broadcast across all 16 columns -> every column of the
// 16x16 accumulator holds the energy tile. 4 waves / block, query row in LDS.
// ---------------------------------------------------------------------------
__global__ __launch_bounds__(128) void energy_wmma_kernel(
    const float* __restrict__ key, const float* __restrict__ query,
    float* __restrict__ energy, int S, int B, int D)
{
    __shared__ float qs[1024];               // D <= 1024 (D = 512 here)
    const int b = blockIdx.y;

    for (int i = threadIdx.x; i < D; i += blockDim.x)
        qs[i] = query[(size_t)b * D + i];
    __syncthreads();

    const int wave = threadIdx.x >> 5;       // 0..3
    const int lane = threadIdx.x & 31;
    const int m    = lane & 15;              // row within 16-tile
    const int h    = lane >> 4;              // 0: K={0,1}, 1: K={2,3}
    const int s0   = (blockIdx.x * 4 + wave) * 16;

    const float* krow = key + (size_t)(s0 + m) * B * D + (size_t)b * D;

    v8f acc = {};
    for (int k0 = 0; k0 < D; k0 += 4) {
        const int kk = k0 + 2 * h;
        // A (16x4 f32): lane m holds A[m, 2h], A[m, 2h+1]
        v2f a = *(const v2f*)(krow + kk);
        // B (4x16 f32): row K striped across lanes; columns replicated with q[k]
        v2f bb = *(const v2f*)(qs + kk);
        acc = __builtin_amdgcn_wmma_f32_16x16x4_f32(
            /*neg_a=*/false, a, /*neg_b=*/false, bb,
            /*c_mod=*/(short)0, acc, /*reuse_a=*/false, /*reuse_b=*/false);
    }

    // D layout: VGPR j -> rows M=j (lanes 0-15) and M=8+j (lanes 16-31),
    // column = lane%16.  Column 0 lanes (m==0) write the 16 energies.
    if (m == 0) {
        float* e = energy + (size_t)b * S + s0 + 8 * h;
        #pragma unroll
        for (int j = 0; j < 8; ++j) e[j] = acc[j];
    }
}

// ---------------------------------------------------------------------------
// Kernel 2: stable softmax over full S, mask s<len, L1-renormalize.
// One block per batch. Writes w straight into the d_out "w" region.
// ---------------------------------------------------------------------------
__global__ __launch_bounds__(256) void softmax_mask_kernel(
    const float* __restrict__ energy, const long long* __restrict__ lens,
    float* __restrict__ w_out, int S)
{
    __shared__ float redA[256];
    __shared__ float redB[256];

    const int b   = blockIdx.x;
    const int tid = threadIdx.x;
    const float* e = energy + (size_t)b * S;
    const long long len = lens[b];

    // --- max over full S ---
    float vmax = -3.402823466e+38f;
    for (int s = tid; s < S; s += 256) vmax = fmaxf(vmax, e[s]);
    redA[tid] = vmax; __syncthreads();
    for (int off = 128; off > 0; off >>= 1) {
        if (tid < off) redA[tid] = fmaxf(redA[tid], redA[tid + off]);
        __syncthreads();
    }
    vmax = redA[0]; __syncthreads();

    // --- full sum and masked sum of exp(e - max) ---
    float sum = 0.f, msum = 0.f;
    for (int s = tid; s < S; s += 256) {
        float p = expf(e[s] - vmax);
        sum += p;
        if ((long long)s < len) msum += p;
    }
    redA[tid] = sum; redB[tid] = msum; __syncthreads();
    for (int off = 128; off > 0; off >>= 1) {
        if (tid < off) { redA[tid] += redA[tid + off]; redB[tid] += redB[tid + off]; }
        __syncthreads();
    }
    const float invZ = 1.f / redA[0];              // softmax denominator
    const float l1   = redB[0] * invZ;             // L1 norm of masked softmax
    const float scale = invZ / fmaxf(l1, 1e-12f);  // combined renormalization

    for (int s = tid; s < S; s += 256) {
        float v = ((long long)s < len) ? expf(e[s] - vmax) * scale : 0.f;
        w_out[(size_t)b * S + s] = v;
    }
}

// ---------------------------------------------------------------------------
// Kernel 3: partial context. grid=(NCHUNK,B), blockDim=D (512).
// Each block streams a 128-row s-chunk of value[.,b,.] fully coalesced
// (512 contiguous floats per s-step) and accumulates a D-vector partial.
// ---------------------------------------------------------------------------
__global__ __launch_bounds__(512) void context_partial_kernel(
    const float* __restrict__ value, const float* __restrict__ w,
    float* __restrict__ partial, int S, int B, int D, int schunk)
{
    __shared__ float wsh[256];                 // schunk <= 256
    const int c = blockIdx.x;
    const int b = blockIdx.y;
    const int d = threadIdx.x;
    const int s0 = c * schunk;

    for (int i = threadIdx.x; i < schunk; i += blockDim.x)
        wsh[i] = w[(size_t)b * S + s0 + i];
    __syncthreads();

    const float* vp = value + (size_t)s0 * B * D + (size_t)b * D + d;
    const size_t stride = (size_t)B * D;

    float acc = 0.f;
    for (int s = 0; s < schunk; ++s)
        acc = fmaf(wsh[s], vp[(size_t)s * stride], acc);

    partial[((size_t)c * B + b) * D + d] = acc;
}

// ---------------------------------------------------------------------------
// Kernel 4: fixed-order reduction of NCHUNK partials -> context (deterministic).
// ---------------------------------------------------------------------------
__global__ __launch_bounds__(256) void reduce_partials_kernel(
    const float* __restrict__ partial, float* __restrict__ out, int n, int nchunk)
{
    const int i = blockIdx.x * blockDim.x + threadIdx.x;
    if (i >= n) return;
    float a = 0.f;
    for (int c = 0; c < nchunk; ++c) a += partial[(size_t)c * n + i];
    out[i] = a;
}

extern "C" void kernel_launch(void* const* d_in, const int* in_sizes, int n_in,
                              void* d_out, int out_size, void* d_ws, size_t ws_size,
                              hipStream_t stream)
{
    const float*     key   = (const float*)d_in[0];      // (S,B,D)
    const float*     value = (const float*)d_in[1];      // (S,B,D)
    const float*     query = (const float*)d_in[2];      // (B,D)
    const long long* lens  = (const long long*)d_in[3];  // (B,) int64

    const int B = in_sizes[3];
    const int D = in_sizes[2] / B;
    const int S = in_sizes[0] / (B * D);

    float* out_ctx = (float*)d_out;                       // (B,D)
    float* out_w   = out_ctx + (size_t)B * D;             // (B,S)

    float* energy  = (float*)d_ws;                        // B*S floats
    const int NCHUNK = 16;
    float* partial = energy + (size_t)B * S;              // NCHUNK*B*D floats

    // 1) energy via f32 WMMA: one wave per 16-row s-tile, 4 waves/block
    dim3 g1(S / 64, B);
    energy_wmma_kernel<<<g1, 128, 0, stream>>>(key, query, energy, S, B, D);

    // 2) softmax + mask + L1 renorm -> w (directly into d_out)
    softmax_mask_kernel<<<B, 256, 0, stream>>>(energy, lens, out_w, S);

    // 3) context partials over s-chunks (coalesced value streaming)
    dim3 g3(NCHUNK, B);
    context_partial_kernel<<<g3, D, 0, stream>>>(value, out_w, partial, S, B, D, S / NCHUNK);

    // 4) deterministic fixed-order reduce -> context
    const int n = B * D;
    reduce_partials_kernel<<<(n + 255) / 256, 256, 0, stream>>>(partial, out_ctx, n, NCHUNK);
}